// SelfGuidedAttention_54245436948846
// MI455X (gfx1250) — compile-verified
//
#include <hip/hip_runtime.h>
#include <hip/hip_bf16.h>
#include <math.h>

// Problem constants (from reference setup_inputs)
#define BATCH 64
#define NPIX  1024      // 32x32
#define HDIM  32
#define DDIM  768
#define KTOP  4         // REGION_NUM - 1
#define BIG   1025      // N + 1
#define INV_SQRT_D 0.03608439182435161f  // 1/sqrt(768)

#define KC        32            // k-columns staged per chunk
#define NCHUNK    (DDIM / KC)   // 24
#define ROWSTRIDE 36            // LDS row pitch (floats): 16B-aligned, bank-conflict-free

typedef __attribute__((ext_vector_type(2))) float v2f;
typedef __attribute__((ext_vector_type(4))) float v4f;
typedef __attribute__((ext_vector_type(8))) float v8f;
typedef __attribute__((ext_vector_type(4))) int   v4i;

#if defined(__gfx1250__) && __has_builtin(__builtin_amdgcn_global_load_async_to_lds_b128)
#define HAVE_ASYNC_LDS 1
#else
#define HAVE_ASYNC_LDS 0
#endif

#if HAVE_ASYNC_LDS
#define AS_GLOBAL __attribute__((address_space(1)))
#define AS_LOCAL  __attribute__((address_space(3)))
#if __has_builtin(__builtin_amdgcn_s_wait_asynccnt)
#define WAIT_ASYNC(n) __builtin_amdgcn_s_wait_asynccnt(n)
#else
#define WAIT_ASYNC(n) asm volatile("s_wait_asynccnt " #n ::: "memory")
#endif
#else
#define WAIT_ASYNC(n) ((void)0)
#endif

// ---------------------------------------------------------------------------
// Stage a 128-row x 32-col f32 tile of z into LDS (row pitch 36 floats).
// Fully coalesced: 256 threads x 4 issues x 16B. Async (ASYNCcnt) when the
// gfx1250 builtin is available, else synchronous B128 load + LDS store.
// ---------------------------------------------------------------------------
__device__ __forceinline__
void stage_chunk(const float* __restrict__ zblk, float* __restrict__ tbuf,
                 int k0, int tid) {
  #pragma unroll
  for (int i = 0; i < 4; ++i) {
    const int e   = i * 256 + tid;       // 0..1023
    const int row = e >> 3;              // 0..127
    const int cg  = (e & 7) * 4;         // col group (floats)
    const float* g = zblk + (size_t)row * DDIM + k0 + cg;
    float*       l = tbuf + row * ROWSTRIDE + cg;
#if HAVE_ASYNC_LDS
    __builtin_amdgcn_global_load_async_to_lds_b128(
        (AS_GLOBAL v4i*)g, (AS_LOCAL v4i*)l, /*offset=*/0, /*cpol=*/0);
#else
    *(v4f*)l = *(const v4f*)g;
#endif
  }
}

// ---------------------------------------------------------------------------
// Kernel 1: sim[b,n] = dot(q[b], z[b,n]) / sqrt(D) via V_WMMA_F32_16X16X4_F32.
// grid = (8, B): blockIdx.x selects a 128-row chunk, blockIdx.y = batch.
// 8 waves; wave w owns rows [w*16, w*16+16). z is staged through a
// double-buffered LDS tile (coalesced/async global loads), WMMA A fragments
// come from LDS (conflict-free ds_load_b64), B = query broadcast across all
// 16 columns so every output column equals sim; lanes 0/16 write rows 0-7/8-15.
// A layout (32-bit A 16x4): lanes 0-15 -> M=lane, K={0,1}; lanes 16-31 -> K={2,3}.
// ---------------------------------------------------------------------------
__global__ __launch_bounds__(256)
void sim_wmma_kernel(const float* __restrict__ z, const float* __restrict__ qz,
                     float* __restrict__ sim) {
  __shared__ float s_q[DDIM];
  __shared__ float s_tile[2][128 * ROWSTRIDE];

  const int b    = blockIdx.y;
  const int tid  = threadIdx.x;
  const int wave = tid >> 5;
  const int lane = tid & 31;

  for (int i = tid; i < DDIM; i += 256) s_q[i] = qz[(size_t)b * DDIM + i];

  const float* zblk = z + ((size_t)b * NPIX + blockIdx.x * 128) * DDIM;

  const int trow = wave * 16 + (lane & 15);   // row within the 128-row block
  const int koff = (lane >> 4) * 2;           // K-pair select (A and B layouts)

  stage_chunk(zblk, s_tile[0], 0, tid);

  v8f c = {};
  for (int j = 0; j < NCHUNK; ++j) {
    if (j + 1 < NCHUNK) {
      stage_chunk(zblk, s_tile[(j + 1) & 1], (j + 1) * KC, tid);
      WAIT_ASYNC(4);             // chunk j complete (j+1 still in flight)
    } else {
      WAIT_ASYNC(0);
    }
    __syncthreads();

    const float* tb = &s_tile[j & 1][trow * ROWSTRIDE + koff];
    const float* qb = &s_q[j * KC + koff];
    #pragma unroll
    for (int kk = 0; kk < KC; kk += 4) {
      v2f a  = *(const v2f*)(tb + kk);       // z rows  (A matrix)
      v2f bq = *(const v2f*)(qb + kk);       // q bcast (B matrix)
      c = __builtin_amdgcn_wmma_f32_16x16x4_f32(
          /*neg_a=*/false, a, /*neg_b=*/false, bq,
          /*c_mod=*/(short)0, c, /*reuse_a=*/false, /*reuse_b=*/false);
    }
    __syncthreads();             // readers done before buffer is overwritten
  }

  // Column 0 of D == sim; lane 0 holds rows 0-7, lane 16 holds rows 8-15.
  if ((lane & 15) == 0) {
    const int rbase = blockIdx.x * 128 + wave * 16 + (lane >> 4) * 8;
    #pragma unroll
    for (int i = 0; i < 8; ++i)
      sim[(size_t)b * NPIX + rbase + i] = c[i] * INV_SQRT_D;
  }
}

// ---------------------------------------------------------------------------
// Kernel 2: per-batch pipeline entirely in LDS. One 256-thread block / batch.
//   normalize -> u8 -> histogram -> Otsu -> mask -> 8-conn CC (Jacobi min
//   propagation) -> segment area/sum -> top-4 by mean -> region masks.
// ---------------------------------------------------------------------------
__global__ __launch_bounds__(256)
void region_pipeline_kernel(const float* __restrict__ sim,
                            float* __restrict__ out) {
  __shared__ float s_sal[NPIX];
  __shared__ int   s_labA[NPIX];
  __shared__ int   s_labB[NPIX];
  __shared__ int   s_hist[256];
  __shared__ float s_w0[256];
  __shared__ float s_mu[256];
  __shared__ float s_red[256];
  __shared__ int   s_redi[256];
  __shared__ float s_area[NPIX + 1];
  __shared__ float s_cand[NPIX + 1];   // ssum, then candidate means
  __shared__ float s_minmax[2];
  __shared__ int   s_thr;
  __shared__ int   s_changed;
  __shared__ int   s_topl[KTOP];
  __shared__ int   s_topsel[KTOP];

  const int b   = blockIdx.x;
  const int tid = threadIdx.x;

  // ---- load sim, init hist ----
  for (int p = tid; p < NPIX; p += 256) s_sal[p] = sim[(size_t)b * NPIX + p];
  s_hist[tid] = 0;
  __syncthreads();

  // ---- block min/max ----
  float lmin = 3.4e38f, lmax = -3.4e38f;
  for (int p = tid; p < NPIX; p += 256) {
    float v = s_sal[p];
    lmin = fminf(lmin, v);
    lmax = fmaxf(lmax, v);
  }
  s_red[tid] = lmin; __syncthreads();
  for (int s = 128; s > 0; s >>= 1) {
    if (tid < s) s_red[tid] = fminf(s_red[tid], s_red[tid + s]);
    __syncthreads();
  }
  if (tid == 0) s_minmax[0] = s_red[0];
  __syncthreads();
  s_red[tid] = lmax; __syncthreads();
  for (int s = 128; s > 0; s >>= 1) {
    if (tid < s) s_red[tid] = fmaxf(s_red[tid], s_red[tid + s]);
    __syncthreads();
  }
  if (tid == 0) s_minmax[1] = s_red[0];
  __syncthreads();

  const float smin = s_minmax[0];
  const float rdenom = 1.0f / (s_minmax[1] - smin + 1e-6f);

  // ---- normalize to sal, histogram u8 ----
  for (int p = tid; p < NPIX; p += 256) {
    float sal = (s_sal[p] - smin) * rdenom;
    s_sal[p] = sal;
    int u = (int)floorf(sal * 255.0f);
    u = u < 0 ? 0 : (u > 255 ? 255 : u);
    atomicAdd(&s_hist[u], 1);
  }
  __syncthreads();

  // ---- Otsu: cumsums (serial, matches jnp.cumsum order), then argmax ----
  if (tid == 0) {
    float w = 0.0f, m = 0.0f;
    for (int t = 0; t < 256; ++t) {
      float pv = (float)s_hist[t] * (1.0f / (float)NPIX);
      w += pv;
      m += pv * (float)t;
      s_w0[t] = w;
      s_mu[t] = m;
    }
  }
  __syncthreads();
  {
    const float mu_t = s_mu[255];
    float w0 = s_w0[tid], mu = s_mu[tid];
    float num = mu_t * w0 - mu;
    s_red[tid] = (num * num) / (w0 * (1.0f - w0) + 1e-12f);
  }
  __syncthreads();
  if (tid == 0) {          // first-max tie-break == jnp.argmax
    int best = 0; float bv = s_red[0];
    for (int t = 1; t < 256; ++t)
      if (s_red[t] > bv) { bv = s_red[t]; best = t; }
    s_thr = best;
  }
  __syncthreads();
  const int thr = s_thr;

  // ---- binary mask + initial labels (flat idx + 1; BIG elsewhere) ----
  for (int p = tid; p < NPIX; p += 256) {
    int u = (int)floorf(s_sal[p] * 255.0f);
    u = u < 0 ? 0 : (u > 255 ? 255 : u);
    s_labA[p] = (u > thr) ? (p + 1) : BIG;
  }
  __syncthreads();

  // ---- 8-connected CC: Jacobi min-label propagation (unique fixed point) ----
  int* cur = s_labA;
  int* nxt = s_labB;
  for (int it = 0; it < NPIX; ++it) {
    if (tid == 0) s_changed = 0;
    __syncthreads();
    for (int p = tid; p < NPIX; p += 256) {
      int old = cur[p];
      int nv = BIG;
      if (old != BIG) {                     // masked pixel (invariant)
        int y = p >> 5, x = p & 31;
        nv = old;
        #pragma unroll
        for (int dy = -1; dy <= 1; ++dy) {
          int yy = y + dy;
          if (yy < 0 || yy >= HDIM) continue;
          #pragma unroll
          for (int dx = -1; dx <= 1; ++dx) {
            int xx = x + dx;
            if (xx < 0 || xx >= HDIM) continue;
            int v = cur[yy * HDIM + xx];
            nv = v < nv ? v : nv;
          }
        }
      }
      nxt[p] = nv;
      if (nv != old) s_changed = 1;
    }
    __syncthreads();
    int* t2 = cur; cur = nxt; nxt = t2;
    if (!s_changed) break;
  }

  // ---- segment sums: area + saliency sum per label (LDS float atomics) ----
  for (int l = tid; l <= NPIX; l += 256) { s_area[l] = 0.0f; s_cand[l] = 0.0f; }
  __syncthreads();
  for (int p = tid; p < NPIX; p += 256) {
    int lab = (cur[p] != BIG) ? cur[p] : 0;
    atomicAdd(&s_area[lab], 1.0f);
    atomicAdd(&s_cand[lab], s_sal[p]);     // s_cand holds ssum for now
  }
  __syncthreads();

  // ---- candidate means ----
  for (int l = tid; l <= NPIX; l += 256) {
    float area = s_area[l];
    float mean = s_cand[l] / fmaxf(area, 1.0f);
    bool valid = (area >= 20.48f) && (l > 0);   // ALPHA * N = 0.02 * 1024
    s_cand[l] = valid ? mean : -INFINITY;
  }
  __syncthreads();

  // ---- top-4 by mean (lowest-index tie-break == lax.top_k) ----
  for (int k = 0; k < KTOP; ++k) {
    float bv = -INFINITY; int bi = 0x7fffffff;
    for (int l = tid; l <= NPIX; l += 256) {
      float v = s_cand[l];
      if (v > bv || (v == bv && l < bi)) { bv = v; bi = l; }
    }
    s_red[tid] = bv; s_redi[tid] = bi;
    __syncthreads();
    if (tid == 0) {
      float BV = s_red[0]; int BI = s_redi[0];
      for (int t = 1; t < 256; ++t)
        if (s_red[t] > BV || (s_red[t] == BV && s_redi[t] < BI)) {
          BV = s_red[t]; BI = s_redi[t];
        }
      s_topl[k]   = BI;
      s_topsel[k] = (BV != -INFINITY) ? 1 : 0;
      s_cand[BI]  = -INFINITY;
    }
    __syncthreads();
  }

  // ---- emit (B, 5, N) region one-hots + background ----
  for (int p = tid; p < NPIX; p += 256) {
    int lab = (cur[p] != BIG) ? cur[p] : 0;
    int any = 0;
    #pragma unroll
    for (int k = 0; k < KTOP; ++k) {
      int r = (s_topsel[k] && lab == s_topl[k]) ? 1 : 0;
      out[((size_t)b * 5 + k) * NPIX + p] = (float)r;
      any |= r;
    }
    out[((size_t)b * 5 + 4) * NPIX + p] = any ? 0.0f : 1.0f;
  }
}

// ---------------------------------------------------------------------------
extern "C" void kernel_launch(void* const* d_in, const int* in_sizes, int n_in,
                              void* d_out, int out_size, void* d_ws, size_t ws_size,
                              hipStream_t stream) {
  const float* z  = (const float*)d_in[0];   // (B, N, D) f32
  const float* qz = (const float*)d_in[1];   // (B, 1, D) f32
  float* out = (float*)d_out;                // (B, 5, N) f32
  float* sim = (float*)d_ws;                 // B*N f32 scratch (256 KB)

  const int B = in_sizes[0] / (NPIX * DDIM);

  dim3 g1(NPIX / 128, B);                    // 8 x 64 = 512 workgroups
  sim_wmma_kernel<<<g1, 256, 0, stream>>>(z, qz, sim);
  region_pipeline_kernel<<<B, 256, 0, stream>>>(sim, out);
}